// HierarchicalAttention_16295105921534
// MI455X (gfx1250) — compile-verified
//
#include <hip/hip_runtime.h>
#include <math.h>

#define WML   64
#define BATCH 64
#define SENTS 40
#define DIM   512
#define NTOK  (SENTS * WML)        // 2560
#define NCHUNK 8
#define ROWS_PER_CHUNK (NTOK / NCHUNK)   // 320
#define ROWS_PER_WAVE  (ROWS_PER_CHUNK / 16) // 20

typedef __attribute__((ext_vector_type(2))) float v2f;
typedef __attribute__((ext_vector_type(8))) float v8f;

__device__ __forceinline__ v8f wmma_f32_16x16x4(v2f a, v2f b, v8f c) {
  // 8-arg pattern: (neg_a, A, neg_b, B, c_mod, C, reuse_a, reuse_b)
  return __builtin_amdgcn_wmma_f32_16x16x4_f32(false, a, false, b, (short)0, c,
                                               false, false);
}

// ---------------------------------------------------------------------------
// K1: Qw = source @ W_word^T ; Qs = source @ W_sent^T   (M=64,N=512,K=512)
// one wave per 16x16 output tile, f32 WMMA, K step = 4
// ---------------------------------------------------------------------------
__global__ __launch_bounds__(32) void proj_gemm_kernel(
    const float* __restrict__ src, const float* __restrict__ Wword,
    const float* __restrict__ Wsent, float* __restrict__ Qw,
    float* __restrict__ Qs) {
  const int m0 = blockIdx.x * 16;
  const int n0 = blockIdx.y * 16;
  const float* W = blockIdx.z ? Wsent : Wword;
  float* Q = blockIdx.z ? Qs : Qw;
  const int lane = threadIdx.x;
  const int row  = lane & 15;
  const int half = lane >> 4;

  v8f acc = {};
  const float* arow = src + (size_t)(m0 + row) * DIM;  // A[m][k]
  const float* brow = W + (size_t)(n0 + row) * DIM;    // B[k][n] = W[n][k]
  for (int k = 0; k < DIM; k += 4) {
    v2f a = *(const v2f*)(arow + k + 2 * half);
    v2f b = *(const v2f*)(brow + k + 2 * half);
    acc = wmma_f32_16x16x4(a, b, acc);
  }
#pragma unroll
  for (int r = 0; r < 8; ++r)
    Q[(size_t)(m0 + r + half * 8) * DIM + n0 + row] = acc[r];
}

// ---------------------------------------------------------------------------
// K2: factor[b,s] = dot(Qs[b,:], sent_bank[s,b,:]) * static_attn[b,s]
// one wave per (b,s)
// ---------------------------------------------------------------------------
__global__ __launch_bounds__(32) void sent_factor_kernel(
    const float* __restrict__ Qs, const float* __restrict__ sent_bank,
    const float* __restrict__ static_attn, float* __restrict__ factor) {
  const int bs = blockIdx.x;
  const int b = bs / SENTS, s = bs % SENTS;
  const int lane = threadIdx.x;
  const float* q = Qs + (size_t)b * DIM;
  const float* v = sent_bank + ((size_t)s * BATCH + b) * DIM;
  float p = 0.f;
  for (int d = lane; d < DIM; d += 32) p += q[d] * v[d];
#pragma unroll
  for (int m = 16; m >= 1; m >>= 1) p += __shfl_xor(p, m, 32);
  if (lane == 0) factor[bs] = p * static_attn[b * SENTS + s];
}

// ---------------------------------------------------------------------------
// K3: single streaming pass over word_bank (online softmax + context accum).
// grid (B, NCHUNK), 512 threads (16 waves). Wave handles 20 rows; lane owns
// dims [lane*16, lane*16+16). Emits raw scores + per-chunk (M, Z, Cpart).
// ---------------------------------------------------------------------------
__global__ __launch_bounds__(512) void flash_word_kernel(
    const float* __restrict__ word_bank, const int* __restrict__ word_lengths,
    const float* __restrict__ Qw, const float* __restrict__ factor,
    float* __restrict__ scores, float* __restrict__ Mc, float* __restrict__ Zc,
    float* __restrict__ Cpart) {
  const int b = blockIdx.x, chunk = blockIdx.y;
  const int tid = threadIdx.x, lane = tid & 31, wave = tid >> 5;

  __shared__ float s_factor[SENTS];
  __shared__ int   s_wlen[SENTS];
  __shared__ float s_m[16], s_z[16];
  __shared__ float s_acc[DIM];

  if (tid < SENTS) {
    s_factor[tid] = factor[b * SENTS + tid];
    s_wlen[tid]   = word_lengths[b * SENTS + tid];
  }

  const float4* qp = (const float4*)(Qw + (size_t)b * DIM + lane * 16);
  const float4 q0 = qp[0], q1 = qp[1], q2 = qp[2], q3 = qp[3];

  float4 c0 = {0.f, 0.f, 0.f, 0.f}, c1 = c0, c2 = c0, c3 = c0;
  float m = -INFINITY, z = 0.f;

  __syncthreads();

  const int nbase = chunk * ROWS_PER_CHUNK;
  for (int r = 0; r < ROWS_PER_WAVE; ++r) {
    const int n = nbase + r * 16 + wave;
    const int s = n >> 6, w = n & 63;
    const float4* rowp = (const float4*)(word_bank +
        (((size_t)w * BATCH + b) * SENTS + s) * DIM + lane * 16);
    const float4 x0 = rowp[0], x1 = rowp[1], x2 = rowp[2], x3 = rowp[3];

    float d = x0.x * q0.x + x0.y * q0.y + x0.z * q0.z + x0.w * q0.w
            + x1.x * q1.x + x1.y * q1.y + x1.z * q1.z + x1.w * q1.w
            + x2.x * q2.x + x2.y * q2.y + x2.z * q2.z + x2.w * q2.w
            + x3.x * q3.x + x3.y * q3.y + x3.z * q3.z + x3.w * q3.w;
#pragma unroll
    for (int mm = 16; mm >= 1; mm >>= 1) d += __shfl_xor(d, mm, 32);

    float sc = d * s_factor[s];
    if (w >= s_wlen[s]) sc = -INFINITY;
    if (lane == 0) scores[(size_t)b * NTOK + n] = sc;

    float wgt;
    if (sc > m) {
      const float rs = __expf(m - sc);  // m == -inf -> 0
      z *= rs;
      c0.x *= rs; c0.y *= rs; c0.z *= rs; c0.w *= rs;
      c1.x *= rs; c1.y *= rs; c1.z *= rs; c1.w *= rs;
      c2.x *= rs; c2.y *= rs; c2.z *= rs; c2.w *= rs;
      c3.x *= rs; c3.y *= rs; c3.z *= rs; c3.w *= rs;
      m = sc;
      wgt = 1.f;
    } else {
      wgt = (m == -INFINITY) ? 0.f : __expf(sc - m);
    }
    z += wgt;
    c0.x += wgt * x0.x; c0.y += wgt * x0.y; c0.z += wgt * x0.z; c0.w += wgt * x0.w;
    c1.x += wgt * x1.x; c1.y += wgt * x1.y; c1.z += wgt * x1.z; c1.w += wgt * x1.w;
    c2.x += wgt * x2.x; c2.y += wgt * x2.y; c2.z += wgt * x2.z; c2.w += wgt * x2.w;
    c3.x += wgt * x3.x; c3.y += wgt * x3.y; c3.z += wgt * x3.z; c3.w += wgt * x3.w;
  }

  if (lane == 0) { s_m[wave] = m; s_z[wave] = z; }
  s_acc[tid] = 0.f;
  __syncthreads();

  float M = -INFINITY;
#pragma unroll
  for (int wv = 0; wv < 16; ++wv) M = fmaxf(M, s_m[wv]);
  const float myscale = (m == -INFINITY) ? 0.f : __expf(m - M);

  const int dbase = lane * 16;
  atomicAdd(&s_acc[dbase + 0],  c0.x * myscale);
  atomicAdd(&s_acc[dbase + 1],  c0.y * myscale);
  atomicAdd(&s_acc[dbase + 2],  c0.z * myscale);
  atomicAdd(&s_acc[dbase + 3],  c0.w * myscale);
  atomicAdd(&s_acc[dbase + 4],  c1.x * myscale);
  atomicAdd(&s_acc[dbase + 5],  c1.y * myscale);
  atomicAdd(&s_acc[dbase + 6],  c1.z * myscale);
  atomicAdd(&s_acc[dbase + 7],  c1.w * myscale);
  atomicAdd(&s_acc[dbase + 8],  c2.x * myscale);
  atomicAdd(&s_acc[dbase + 9],  c2.y * myscale);
  atomicAdd(&s_acc[dbase + 10], c2.z * myscale);
  atomicAdd(&s_acc[dbase + 11], c2.w * myscale);
  atomicAdd(&s_acc[dbase + 12], c3.x * myscale);
  atomicAdd(&s_acc[dbase + 13], c3.y * myscale);
  atomicAdd(&s_acc[dbase + 14], c3.z * myscale);
  atomicAdd(&s_acc[dbase + 15], c3.w * myscale);
  __syncthreads();

  const int idx = b * NCHUNK + chunk;
  Cpart[(size_t)idx * DIM + tid] = s_acc[tid];
  if (tid == 0) {
    float Z = 0.f;
#pragma unroll
    for (int wv = 0; wv < 16; ++wv)
      Z += s_z[wv] * ((s_m[wv] == -INFINITY) ? 0.f : __expf(s_m[wv] - M));
    Mc[idx] = M;
    Zc[idx] = Z;
  }
}

// ---------------------------------------------------------------------------
// K4: per-b combine: global softmax stats, final c[b,:], align_vectors out.
// ---------------------------------------------------------------------------
__global__ __launch_bounds__(512) void combine_kernel(
    const float* __restrict__ scores, const float* __restrict__ Mc,
    const float* __restrict__ Zc, const float* __restrict__ Cpart,
    float* __restrict__ Cfinal, float* __restrict__ out_av) {
  const int b = blockIdx.x, tid = threadIdx.x;

  float Ms[NCHUNK], Zs[NCHUNK], scl[NCHUNK];
  float Mb = -INFINITY;
#pragma unroll
  for (int ch = 0; ch < NCHUNK; ++ch) {
    Ms[ch] = Mc[b * NCHUNK + ch];
    Zs[ch] = Zc[b * NCHUNK + ch];
    Mb = fmaxf(Mb, Ms[ch]);
  }
  float Zb = 0.f;
#pragma unroll
  for (int ch = 0; ch < NCHUNK; ++ch) {
    scl[ch] = (Ms[ch] == -INFINITY) ? 0.f : __expf(Ms[ch] - Mb);
    Zb += Zs[ch] * scl[ch];
  }
  const float inv = 1.f / Zb;

  float c = 0.f;
#pragma unroll
  for (int ch = 0; ch < NCHUNK; ++ch)
    c += Cpart[(size_t)(b * NCHUNK + ch) * DIM + tid] * scl[ch];
  Cfinal[(size_t)b * DIM + tid] = c * inv;

  for (int n = tid; n < NTOK; n += 512) {
    const float sc = scores[(size_t)b * NTOK + n];
    const float av = (sc == -INFINITY) ? 0.f : __expf(sc - Mb) * inv;
    out_av[(size_t)b * NTOK + n] = av + 1e-20f;
  }
}

// ---------------------------------------------------------------------------
// K5: attn_h = tanh([c, source] @ W_out^T)  (M=64,N=512,K=1024) via f32 WMMA
// ---------------------------------------------------------------------------
__global__ __launch_bounds__(32) void out_gemm_kernel(
    const float* __restrict__ Cfinal, const float* __restrict__ src,
    const float* __restrict__ Wout, float* __restrict__ out) {
  const int m0 = blockIdx.x * 16;
  const int n0 = blockIdx.y * 16;
  const int lane = threadIdx.x;
  const int row  = lane & 15;
  const int half = lane >> 4;

  v8f acc = {};
  const int am = m0 + row;
  const float* wrow = Wout + (size_t)(n0 + row) * (2 * DIM);
  for (int k = 0; k < 2 * DIM; k += 4) {
    const int ka = k + 2 * half;  // ka and ka+1 stay on the same side of DIM
    const float* ap = (ka < DIM) ? (Cfinal + (size_t)am * DIM + ka)
                                 : (src + (size_t)am * DIM + (ka - DIM));
    v2f a = *(const v2f*)ap;
    v2f bfrag = *(const v2f*)(wrow + ka);
    acc = wmma_f32_16x16x4(a, bfrag, acc);
  }
#pragma unroll
  for (int r = 0; r < 8; ++r)
    out[(size_t)(m0 + r + half * 8) * DIM + n0 + row] = tanhf(acc[r]);
}

// ---------------------------------------------------------------------------
extern "C" void kernel_launch(void* const* d_in, const int* in_sizes, int n_in,
                              void* d_out, int out_size, void* d_ws,
                              size_t ws_size, hipStream_t stream) {
  (void)in_sizes; (void)n_in; (void)out_size; (void)ws_size;
  const float* source       = (const float*)d_in[0];
  const float* word_bank    = (const float*)d_in[1];
  const int*   word_lengths = (const int*)d_in[2];
  const float* sent_bank    = (const float*)d_in[3];
  /* d_in[4] sent_lengths unused by the forward math */
  const float* static_attn  = (const float*)d_in[5];
  const float* W_word       = (const float*)d_in[6];
  const float* W_sent       = (const float*)d_in[7];
  const float* W_out        = (const float*)d_in[8];
  float* out = (float*)d_out;  // [B*D attn_h | B*NTOK align_vectors]

  float* ws     = (float*)d_ws;
  float* Qw     = ws;                          // B*D
  float* Qs     = Qw + BATCH * DIM;            // B*D
  float* factor = Qs + BATCH * DIM;            // B*S
  float* scores = factor + BATCH * SENTS;      // B*NTOK
  float* Mc     = scores + BATCH * NTOK;       // B*NCHUNK
  float* Zc     = Mc + BATCH * NCHUNK;         // B*NCHUNK
  float* Cpart  = Zc + BATCH * NCHUNK;         // B*NCHUNK*D
  float* Cfinal = Cpart + BATCH * NCHUNK * DIM;// B*D

  proj_gemm_kernel<<<dim3(BATCH / 16, DIM / 16, 2), 32, 0, stream>>>(
      source, W_word, W_sent, Qw, Qs);
  sent_factor_kernel<<<BATCH * SENTS, 32, 0, stream>>>(
      Qs, sent_bank, static_attn, factor);
  flash_word_kernel<<<dim3(BATCH, NCHUNK), 512, 0, stream>>>(
      word_bank, word_lengths, Qw, factor, scores, Mc, Zc, Cpart);
  combine_kernel<<<BATCH, 512, 0, stream>>>(
      scores, Mc, Zc, Cpart, Cfinal, out + BATCH * DIM);
  out_gemm_kernel<<<dim3(BATCH / 16, DIM / 16), 32, 0, stream>>>(
      Cfinal, source, W_out, out);
}